// HungarianMatcher_20933670600915
// MI455X (gfx1250) — compile-verified
//
#include <hip/hip_runtime.h>
#include <hip/hip_bf16.h>

// Problem constants (match reference)
#define BSV   16
#define NQV   900
#define NTV   100
#define NTTOT (BSV * NTV)        // 1600 targets total
#define NPRED (BSV * NQV)        // 14400 predictions
#define C_ELEMS ((size_t)NPRED * NTTOT)  // 23,040,000

#define W_BBOX  5.0f
#define W_GIOU  2.0f
#define W_HUMAN 1.0f

typedef unsigned int u32x4 __attribute__((ext_vector_type(4)));
typedef int          i32x8 __attribute__((ext_vector_type(8)));
typedef int          i32x4 __attribute__((ext_vector_type(4)));

#if defined(__HIP_DEVICE_COMPILE__) && __has_builtin(__builtin_amdgcn_tensor_load_to_lds) && __has_builtin(__builtin_amdgcn_s_wait_tensorcnt)
#define HAVE_TDM 1
#else
#define HAVE_TDM 0
#endif

#if HAVE_TDM
// Issue a 1-D TDM copy of `nelem` f32 from global `gsrc` into LDS byte address
// `lds_addr`. Descriptor per CDNA5 ISA section 8 (async_tensor):
//   group0: count=1 | lds_addr | global_addr[56:0] | type=2
//   group1: data_size=4B(code 2), tensor_dim0=nelem, tile_dim0=nelem,
//           tensor_dim1=1, higher dims unused, no padding / iterate / multicast
// This toolchain's builtin takes 6 args (clang-23 / therock-10.0 form):
//   (u32x4 g0, i32x8 g1, i32x4 g2, i32x4 g3, i32x8 g4, i32 cpol)
__device__ __forceinline__ void tdm_load_f32_1d(const float* gsrc,
                                                unsigned int lds_addr,
                                                unsigned int nelem) {
  unsigned long long ga = (unsigned long long)(uintptr_t)gsrc;
  u32x4 g0;
  g0[0] = 1u;                                   // count=1, is_restore=0, gather off
  g0[1] = lds_addr;                             // LDS byte address
  g0[2] = (unsigned int)(ga & 0xFFFFFFFFu);     // global_addr[31:0]
  g0[3] = (unsigned int)((ga >> 32) & 0x1FFFFFFu) | (2u << 30); // addr[56:32] | type=2
  i32x8 g1;
  g1[0] = (int)(2u << 16);                      // wg_mask=0, data_size=4B, no flags
  g1[1] = (int)((nelem & 0xFFFFu) << 16);       // atomic_addr=0 | tensor_dim0[15:0]
  g1[2] = (int)((nelem >> 16) | (1u << 16));    // tensor_dim0[31:16] | tensor_dim1[15:0]=1
  g1[3] = (int)((nelem & 0xFFFFu) << 16);       // tensor_dim1[31:16]=0 | tile_dim0
  g1[4] = 0;                                    // tile_dim1=0, tile_dim2=0 (unused)
  g1[5] = (int)nelem;                           // tensor_dim0_stride[31:0]
  g1[6] = 0;                                    // stride0[47:32] | tensor_dim1_stride[15:0]
  g1[7] = 0;                                    // tensor_dim1_stride[47:16]
  i32x4 g2 = {0, 0, 0, 0};                      // dims 2/3 unused
  i32x4 g3 = {0, 0, 0, 0};                      // dim 4 unused
  i32x8 g4 = {0, 0, 0, 0, 0, 0, 0, 0};          // extra group (unused, 1-D tile)
  __builtin_amdgcn_tensor_load_to_lds(g0, g1, g2, g3, g4, 0);
}
#endif

// ---------------------------------------------------------------------------
// Kernel 1: cost matrix.  One block = 8 pred rows x all 1600 targets.
// Targets staged in LDS via the Tensor Data Mover (25.6 KB, one DMA per block).
// Stores along the target axis are fully coalesced B32.
// ---------------------------------------------------------------------------
__global__ __launch_bounds__(256) void cost_kernel(
    const float* __restrict__ pred,    // [NPRED][4] cxcywh
    const float* __restrict__ logits,  // [NPRED][2]
    const float* __restrict__ tgt,     // [NTTOT][4] cxcywh
    float* __restrict__ C)             // [NPRED][NTTOT]
{
  __shared__ float sT[NTTOT * 4];
  const int tid = threadIdx.x;

#if HAVE_TDM
  if (tid < 32) {
    tdm_load_f32_1d(tgt, (unsigned int)(uintptr_t)&sT[0], NTTOT * 4);
    __builtin_amdgcn_s_wait_tensorcnt(0);   // TENSORcnt == 0 -> LDS data visible
  }
  __syncthreads();
#else
  for (int k = tid; k < NTTOT * 4; k += 256) sT[k] = tgt[k];
  __syncthreads();
#endif

  const int row0 = blockIdx.x * 8;
#pragma unroll
  for (int r = 0; r < 8; ++r) {
    const int q = row0 + r;
    if (q >= NPRED) break;
    // Uniform per-row pred data -> scalar loads
    const float pcx = pred[q * 4 + 0];
    const float pcy = pred[q * 4 + 1];
    const float pw  = pred[q * 4 + 2];
    const float ph  = pred[q * 4 + 3];
    const float px1 = pcx - 0.5f * pw, py1 = pcy - 0.5f * ph;
    const float px2 = pcx + 0.5f * pw, py2 = pcy + 0.5f * ph;
    const float parea = pw * ph;
    const float l0 = logits[q * 2 + 0];
    const float l1 = logits[q * 2 + 1];
    // cost_human = -softmax(l)[0] = -1/(1+exp(l1-l0))
    const float chuman = -1.0f / (1.0f + expf(l1 - l0));

    float* __restrict__ Crow = C + (size_t)q * NTTOT;
    for (int j = tid; j < NTTOT; j += 256) {
      const float4 t = ((const float4*)sT)[j];
      const float tcx = t.x, tcy = t.y, tw = t.z, th = t.w;
      // L1 on cxcywh
      const float cb = fabsf(pcx - tcx) + fabsf(pcy - tcy) +
                       fabsf(pw - tw)   + fabsf(ph - th);
      // GIoU on xyxy
      const float tx1 = tcx - 0.5f * tw, ty1 = tcy - 0.5f * th;
      const float tx2 = tcx + 0.5f * tw, ty2 = tcy + 0.5f * th;
      const float tarea = tw * th;
      const float iw = fminf(px2, tx2) - fmaxf(px1, tx1);
      const float ih = fminf(py2, ty2) - fmaxf(py1, ty1);
      const float inter = fmaxf(iw, 0.0f) * fmaxf(ih, 0.0f);
      const float uni = parea + tarea - inter;
      const float iou = inter / uni;
      const float ew = fmaxf(fmaxf(px2, tx2) - fminf(px1, tx1), 0.0f);
      const float eh = fmaxf(fmaxf(py2, ty2) - fminf(py1, ty1), 0.0f);
      const float earea = ew * eh;
      const float giou = iou - (earea - uni) / earea;
      Crow[j] = W_BBOX * cb + W_GIOU * (-giou) + W_HUMAN * chuman;
    }
  }
}

// ---------------------------------------------------------------------------
// Kernel 2: Jonker-Volgenant LSA, one workgroup per batch.
// Solves the transposed problem (n=100 target rows, m=900 pred cols) in
// double precision to mirror the reference.  Column scans and the delta
// updates are partitioned across 256 threads; the argmin uses an LDS tree
// reduction with smallest-index tie-break (== np.argmin semantics).
// cost_T[i][j] = C[b][j][b*100 + i]  (strided reads; whole C fits in 192MB L2)
// ---------------------------------------------------------------------------
__global__ __launch_bounds__(256) void lsa_kernel(
    const float* __restrict__ C,
    float* __restrict__ rows_out,   // [BSV][NTV] pred indices
    float* __restrict__ cols_out)   // [BSV][NTV] target indices
{
  const int b   = blockIdx.x;
  const int tid = threadIdx.x;
  const int n = NTV, m = NQV;
  const float* __restrict__ Cb = C + (size_t)b * NQV * NTTOT + (size_t)b * NTV;
  // cost_T[i][j] == Cb[j*NTTOT + i]   (i,j zero-based)

  __shared__ double u[NTV + 1];
  __shared__ double v[NQV + 1];
  __shared__ double minv[NQV + 1];
  __shared__ int    way[NQV + 1];
  __shared__ int    p[NQV + 1];
  __shared__ unsigned char used[NQV + 1];
  __shared__ double redv[256];
  __shared__ int    redj[256];
  __shared__ int    s_j0;
  __shared__ double s_delta;

  for (int j = tid; j <= m; j += 256) { v[j] = 0.0; p[j] = 0; }
  for (int i = tid; i <= n; i += 256) u[i] = 0.0;
  __syncthreads();

  for (int i = 1; i <= n; ++i) {
    if (tid == 0) { p[0] = i; s_j0 = 0; }
    for (int j = tid; j <= m; j += 256) { minv[j] = __builtin_inf(); used[j] = 0; }
    __syncthreads();

    for (;;) {
      const int j0 = s_j0;
      if (tid == 0) used[j0] = 1;
      __syncthreads();
      const int    i0   = p[j0];
      const double u_i0 = u[i0];
      const float* __restrict__ crow = Cb + (i0 - 1);  // element j -> crow[(j-1)*NTTOT]

      double bestv = __builtin_inf();
      int    bestj = m + 1;
      for (int j = tid + 1; j <= m; j += 256) {
        if (!used[j]) {
          const double cur = (double)crow[(size_t)(j - 1) * NTTOT] - u_i0 - v[j];
          if (cur < minv[j]) { minv[j] = cur; way[j] = j0; }
          const double mv = minv[j];
          if (mv < bestv) { bestv = mv; bestj = j; }  // j ascending -> first min
        }
      }
      redv[tid] = bestv; redj[tid] = bestj;
      __syncthreads();
      for (int s = 128; s > 0; s >>= 1) {
        if (tid < s) {
          const double ov = redv[tid + s]; const int oj = redj[tid + s];
          if (ov < redv[tid] || (ov == redv[tid] && oj < redj[tid])) {
            redv[tid] = ov; redj[tid] = oj;
          }
        }
        __syncthreads();
      }
      if (tid == 0) { s_delta = redv[0]; s_j0 = redj[0]; }
      __syncthreads();
      const double delta = s_delta;
      const int    j1    = s_j0;
      // u[p[used]] += delta ; v[used] -= delta ; minv[~used] -= delta
      for (int j = tid; j <= m; j += 256) {
        if (used[j]) { u[p[j]] += delta; v[j] -= delta; }
        else         { minv[j] -= delta; }
      }
      __syncthreads();
      if (p[j1] == 0) break;   // uniform: p is stable inside this loop
    }

    if (tid == 0) {            // augmenting path flip (cheap, serial)
      int j0 = s_j0;
      while (j0) { const int j1 = way[j0]; p[j0] = p[j1]; j0 = j1; }
    }
    __syncthreads();
  }

  // Assigned columns in ascending j == pairs sorted by pred index (reference
  // order after its transpose-back + argsort(rows)).
  if (tid == 0) {
    int k = 0;
    for (int j = 1; j <= m; ++j) {
      if (p[j] != 0) {
        rows_out[b * NTV + k] = (float)(j - 1);    // pred index 0..899
        cols_out[b * NTV + k] = (float)(p[j] - 1); // target index 0..99
        ++k;
      }
    }
  }
}

extern "C" void kernel_launch(void* const* d_in, const int* in_sizes, int n_in,
                              void* d_out, int out_size, void* d_ws, size_t ws_size,
                              hipStream_t stream) {
  (void)in_sizes; (void)n_in; (void)d_ws; (void)ws_size; (void)out_size;
  const float* pred   = (const float*)d_in[0];  // (16,900,4)
  const float* logits = (const float*)d_in[1];  // (16,900,2)
  const float* tgt    = (const float*)d_in[2];  // (1600,4)
  float* C    = (float*)d_out;
  float* rows = C + C_ELEMS;
  float* cols = rows + (size_t)BSV * NTV;

  const int blocksA = (NPRED + 7) / 8;          // 1800
  cost_kernel<<<blocksA, 256, 0, stream>>>(pred, logits, tgt, C);
  lsa_kernel<<<BSV, 256, 0, stream>>>(C, rows, cols);
}